// Complex_CA_30021821399811
// MI455X (gfx1250) — compile-verified
//
#include <hip/hip_runtime.h>
#include <hip/hip_bf16.h>

typedef __attribute__((ext_vector_type(2))) float v2f;
typedef __attribute__((ext_vector_type(8))) float v8f;

#define NPIX 289           // 17*17
#define XP   19            // padded plane edge
#define XPA  (XP*XP)       // 361

__device__ __constant__ float KDX[9] = {
    -0.125f, 0.f, 0.125f,
    -0.25f,  0.f, 0.25f,
    -0.125f, 0.f, 0.125f};
__device__ __constant__ float KDY[9] = {
    -0.125f, -0.25f, -0.125f,
     0.f,     0.f,    0.f,
     0.125f,  0.25f,  0.125f};
__device__ __constant__ float K5[25] = {
    0.f,    0.125f, 0.25f, 0.125f, 0.f,
    0.125f, 0.25f,  0.5f,  0.25f,  0.125f,
    0.25f,  0.5f,   1.0f,  0.5f,   0.25f,
    0.125f, 0.25f,  0.5f,  0.25f,  0.125f,
    0.f,    0.125f, 0.25f, 0.125f, 0.f};

__device__ __forceinline__ float conv3(const float* plane, int base, const float* k) {
    // base = (pr)*XP + pc  (top-left of 3x3 window in padded plane)
    float s = 0.f;
#pragma unroll
    for (int i = 0; i < 3; ++i)
#pragma unroll
        for (int j = 0; j < 3; ++j)
            s = fmaf(k[i * 3 + j], plane[base + i * XP + j], s);
    return s;
}

__device__ __forceinline__ float maxpool3(const float* plane, int base) {
    float m = -1e30f;
#pragma unroll
    for (int i = 0; i < 3; ++i)
#pragma unroll
        for (int j = 0; j < 3; ++j)
            m = fmaxf(m, plane[base + i * XP + j]);
    return m;
}

__global__ __launch_bounds__(256)
void Complex_CA_kernel(const float* __restrict__ cell, const float* __restrict__ food,
                       const float* __restrict__ w3, const float* __restrict__ b3,
                       const float* __restrict__ w4, const float* __restrict__ b4,
                       const int* __restrict__ steps_p,
                       float* __restrict__ out, int B)
{
    const int b    = blockIdx.x;
    const int tid  = threadIdx.x;
    const int lane = tid & 31;
    const int wave = tid >> 5;
    const int n16  = lane & 15;
    const bool hi  = lane >= 16;

    __shared__ float sx[4 * XPA];     // padded state, zero halo
    __shared__ float sxn[4 * XPA];    // padded xn, zero halo
    __shared__ float sscent[NPIX];
    __shared__ float spre[NPIX];
    __shared__ float sfood[NPIX];
    __shared__ float shb[8][16][17];  // per-wave h tile, padded row -> no bank conflicts
    __shared__ float s_red[2];

    // ---- init: zero halos, stage inputs, one-time 5x5 scent conv ----
    for (int i = tid; i < 4 * XPA; i += 256) { sx[i] = 0.f; sxn[i] = 0.f; }
    for (int i = tid; i < NPIX; i += 256) sfood[i] = food[(size_t)b * NPIX + i];
    if (tid == 0) { s_red[0] = 0.f; s_red[1] = 0.f; }
    __syncthreads();
    for (int i = tid; i < 4 * NPIX; i += 256) {
        int ch = i / NPIX, p = i - ch * NPIX;
        int r = p / 17, c = p % 17;
        sx[ch * XPA + (r + 1) * XP + (c + 1)] = cell[(size_t)b * 4 * NPIX + i];
    }
    for (int p = tid; p < NPIX; p += 256) {
        int r = p / 17, c = p % 17;
        float s = 0.f;
#pragma unroll
        for (int dr = -2; dr <= 2; ++dr)
#pragma unroll
            for (int dc = -2; dc <= 2; ++dc) {
                int rr = r + dr, cc = c + dc;
                float v = (rr >= 0 && rr < 17 && cc >= 0 && cc < 17) ? sfood[rr * 17 + cc] : 0.f;
                s = fmaf(K5[(dr + 2) * 5 + (dc + 2)], v, s);
            }
        sscent[p] = s;
    }

    // ---- preload weights into WMMA B-layout registers (16x16x4 f32) ----
    // B is KxN: lanes 0-15 hold K=0 (vgpr0) / K=1 (vgpr1), lanes 16-31 hold K=2/K=3, N = lane&15
    v2f w3b[3], w4b[4];
#pragma unroll
    for (int c = 0; c < 3; ++c) {
        w3b[c].x = w3[n16 * 12 + 4 * c + (hi ? 2 : 0)];
        w3b[c].y = w3[n16 * 12 + 4 * c + (hi ? 3 : 1)];
    }
#pragma unroll
    for (int c = 0; c < 4; ++c) {
        w4b[c].x = (n16 < 4) ? w4[n16 * 16 + 4 * c + (hi ? 2 : 0)] : 0.f;
        w4b[c].y = (n16 < 4) ? w4[n16 * 16 + 4 * c + (hi ? 3 : 1)] : 0.f;
    }
    const float b3n = b3[n16];
    const float b4n = (n16 < 4) ? b4[n16] : 0.f;
    const int c0 = hi ? 2 : 0;                 // K-half select per lane half

    const int nsteps = *steps_p;
    __syncthreads();

    // ---- 32-step recurrence, fully resident in LDS ----
    for (int s = 0; s < nsteps; ++s) {
        // ch3 <- scent
        for (int p = tid; p < NPIX; p += 256) {
            int r = p / 17, c = p % 17;
            sx[3 * XPA + (r + 1) * XP + (c + 1)] = sscent[p];
        }
        __syncthreads();
        // pre gate on ch0
        for (int p = tid; p < NPIX; p += 256) {
            int r = p / 17, c = p % 17;
            spre[p] = (maxpool3(&sx[0], r * XP + c) > 0.1f) ? 1.f : 0.f;
        }
        __syncthreads();

        // 19 tiles of 16 pixels across 8 waves (wave-uniform loop: EXEC all-1s at WMMA)
        for (int t = wave; t < 19; t += 8) {
            const int   pix   = t * 16 + n16;       // A layout: M = lane&15 in both halves
            const float fmask = (pix < NPIX) ? 1.f : 0.f;  // branchless OOB handling
            const int   pixc  = (pix < NPIX) ? pix : (NPIX - 1);
            const int   pr    = pixc / 17;
            const int   pc    = pixc - pr * 17;
            const int   bctr  = (pr + 1) * XP + (pc + 1);  // center in padded plane
            const int   btl   = pr * XP + pc;              // 3x3 window top-left

            v8f h = {};
            v2f a;
            // K chunk 0: raw channels x[c0], x[c0+1]  (unconditional LDS loads, masked)
            a.x = fmask * sx[c0 * XPA + bctr];
            a.y = fmask * sx[(c0 + 1) * XPA + bctr];
            h = __builtin_amdgcn_wmma_f32_16x16x4_f32(false, a, false, w3b[0], (short)0, h, false, false);
            // K chunk 1: depthwise dX
            a.x = fmask * conv3(&sx[c0 * XPA], btl, KDX);
            a.y = fmask * conv3(&sx[(c0 + 1) * XPA], btl, KDX);
            h = __builtin_amdgcn_wmma_f32_16x16x4_f32(false, a, false, w3b[1], (short)0, h, false, false);
            // K chunk 2: depthwise dY
            a.x = fmask * conv3(&sx[c0 * XPA], btl, KDY);
            a.y = fmask * conv3(&sx[(c0 + 1) * XPA], btl, KDY);
            h = __builtin_amdgcn_wmma_f32_16x16x4_f32(false, a, false, w3b[2], (short)0, h, false, false);

            // bias + ReLU on accumulator (C/D layout: vgpr r -> M=r(+8), N=lane&15)
            const int mrow = hi ? 8 : 0;
#pragma unroll
            for (int r = 0; r < 8; ++r) h[r] = fmaxf(h[r] + b3n, 0.f);
            // D-layout -> A-layout relayout via per-wave LDS tile
#pragma unroll
            for (int r = 0; r < 8; ++r) shb[wave][r + mrow][n16] = h[r];

            v8f d = {};
#pragma unroll
            for (int c = 0; c < 4; ++c) {
                v2f a2;
                a2.x = shb[wave][n16][4 * c + (hi ? 2 : 0)];
                a2.y = shb[wave][n16][4 * c + (hi ? 3 : 1)];
                d = __builtin_amdgcn_wmma_f32_16x16x4_f32(false, a2, false, w4b[c], (short)0, d, false, false);
            }
            // xn = x + d + b4 (un-gated), only N = out-channel < 4 is real
            if (n16 < 4) {
#pragma unroll
                for (int r = 0; r < 8; ++r) {
                    int p2 = t * 16 + r + mrow;
                    if (p2 < NPIX) {
                        int rr = p2 / 17, cc = p2 - (p2 / 17) * 17;
                        int idx = n16 * XPA + (rr + 1) * XP + (cc + 1);
                        sxn[idx] = sx[idx] + d[r] + b4n;
                    }
                }
            }
        }
        __syncthreads();

        // post gate, life mask, clip -> next state
        for (int p = tid; p < NPIX; p += 256) {
            int r = p / 17, c = p % 17;
            float post = maxpool3(&sxn[0], r * XP + c);
            float life = (spre[p] != 0.f && post > 0.1f) ? 1.f : 0.f;
#pragma unroll
            for (int ch = 0; ch < 4; ++ch) {
                int idx = ch * XPA + (r + 1) * XP + (c + 1);
                float v = sxn[idx] * life;
                sx[idx] = fminf(fmaxf(v, -10.f), 10.f);
            }
        }
        __syncthreads();
    }

    // ---- outputs: cell | food | living_count | living_above ----
    const size_t o_cell = (size_t)b * 4 * NPIX;
    const size_t O_food = (size_t)B * 4 * NPIX;
    const size_t O_lc   = O_food + (size_t)B * NPIX;
    const size_t O_la   = O_lc + (size_t)B;
    float psum = 0.f, pcnt = 0.f;
    for (int p = tid; p < NPIX; p += 256) {
        int r = p / 17, c = p % 17;
#pragma unroll
        for (int ch = 0; ch < 4; ++ch)
            out[o_cell + (size_t)ch * NPIX + p] = sx[ch * XPA + (r + 1) * XP + (c + 1)];
        out[O_food + (size_t)b * NPIX + p] = sfood[p];
        float v0 = sx[(r + 1) * XP + (c + 1)];
        psum += v0;
        pcnt += (v0 > 0.1f) ? 1.f : 0.f;
    }
    atomicAdd(&s_red[0], psum);
    atomicAdd(&s_red[1], pcnt);
    __syncthreads();
    if (tid == 0) {
        out[O_lc + b] = s_red[0];
        out[O_la + b] = s_red[1];
    }
}

extern "C" void kernel_launch(void* const* d_in, const int* in_sizes, int n_in,
                              void* d_out, int out_size, void* d_ws, size_t ws_size,
                              hipStream_t stream) {
    const float* cell  = (const float*)d_in[0];
    const float* food  = (const float*)d_in[1];
    const float* w3    = (const float*)d_in[2];
    const float* b3    = (const float*)d_in[3];
    const float* w4    = (const float*)d_in[4];
    const float* b4    = (const float*)d_in[5];
    const int*   steps = (const int*)d_in[6];
    float* out = (float*)d_out;
    int B = in_sizes[0] / (4 * NPIX);
    Complex_CA_kernel<<<B, 256, 0, stream>>>(cell, food, w3, b3, w4, b4, steps, out, B);
}